// MatchingLayer_34308198761239
// MI455X (gfx1250) — compile-verified
//
#include <hip/hip_runtime.h>

// ---------------------------------------------------------------------------
// Batched QP interior-point solver for the 16x16 soft-matching layer.
//   min 0.5*EPS*||z||^2 + p^T z  s.t. row/col sums <= 1, z >= 0,  p = -x
// One wave32 per batch element. The Newton system M dz = rhs is solved via
// Woodbury + Schur: only a 16x16 SPD solve remains. The 16x16x16 product
// E^T (A^{-1} E) uses V_WMMA_F32_16X16X4_F32 (4 chained K=4 slices).
// All 16x16 quantities live in the CDNA5 WMMA f32 C/D register layout:
//   reg r (0..7), lane l:  element (row = r + 8*(l>>4), col = l&15)
// The 32 row/col constraints map onto the 32 lanes (lane<16: row i=lane,
// lane>=16: col j=lane-16).
//
// Round-3 changes: all cross-lane movement moved off the LDS pipe:
//  * row reductions  -> v_mov_dpp ROW_ROR rotate-reduce (VALU only)
//  * half exchange   -> v_permlanex16 (VALU only)
//  * const-lane bcast-> v_readlane pairs + cndmask
// leaving (ideally) zero ds_bpermute / s_wait_dscnt in the iteration body.
// ---------------------------------------------------------------------------

typedef float v2f __attribute__((ext_vector_type(2)));
typedef float v8f __attribute__((ext_vector_type(8)));

#define QP_EPS  1e-4f
#define QP_SIG  0.1f
#define QP_IT   25

// ---- cross-lane primitives --------------------------------------------------

__device__ __forceinline__ float rdlane(float v, int l) {
  return __int_as_float(__builtin_amdgcn_readlane(__float_as_int(v), l));
}

#if __has_builtin(__builtin_amdgcn_update_dpp)
template <int CTRL>
__device__ __forceinline__ float dppmov(float v) {
  return __int_as_float(__builtin_amdgcn_update_dpp(
      0, __float_as_int(v), CTRL, 0xF, 0xF, true));
}
#define QP_HAVE_DPP 1
#endif

// exchange with the other 16-lane row (same position): xor-16
__device__ __forceinline__ float xhalf(float v) {
#if __has_builtin(__builtin_amdgcn_permlanex16)
  int i = __float_as_int(v);
  return __int_as_float(__builtin_amdgcn_permlanex16(
      i, i, 0x76543210, 0xFEDCBA98, false, false));
#else
  return __shfl_xor(v, 16, 32);
#endif
}

// full sum over the 16 lanes of each row (result in every lane of the row)
__device__ __forceinline__ float half_rowsum(float v) {
#ifdef QP_HAVE_DPP
  v += dppmov<0x121>(v);   // row_ror:1
  v += dppmov<0x122>(v);   // row_ror:2
  v += dppmov<0x124>(v);   // row_ror:4
  v += dppmov<0x128>(v);   // row_ror:8
#else
  v += __shfl_xor(v, 1, 32); v += __shfl_xor(v, 2, 32);
  v += __shfl_xor(v, 4, 32); v += __shfl_xor(v, 8, 32);
#endif
  return v;
}

__device__ __forceinline__ float wave_sum(float v) {
  v = half_rowsum(v);
  return v + xhalf(v);
}

__device__ __forceinline__ float wave_min(float v) {
#ifdef QP_HAVE_DPP
  v = fminf(v, dppmov<0x121>(v));
  v = fminf(v, dppmov<0x122>(v));
  v = fminf(v, dppmov<0x124>(v));
  v = fminf(v, dppmov<0x128>(v));
#else
  v = fminf(v, __shfl_xor(v, 1, 32)); v = fminf(v, __shfl_xor(v, 2, 32));
  v = fminf(v, __shfl_xor(v, 4, 32)); v = fminf(v, __shfl_xor(v, 8, 32));
#endif
  return fminf(v, xhalf(v));
}

// fast reciprocal: v_rcp_f32 + one Newton-Raphson step (~0.5 ulp)
__device__ __forceinline__ float frcp(float x) {
  float r = __builtin_amdgcn_rcpf(x);
  r = r * __builtin_fmaf(-x, r, 2.0f);
  return r;
}

// broadcast rc-layout row entry i = r (h=0) / r+8 (h=1); lanes are constant
__device__ __forceinline__ float bcast_row(float v_rc, int r, int h) {
  float a = rdlane(v_rc, r);
  float b = rdlane(v_rc, r + 8);
  return h ? b : a;
}

// broadcast rc-layout col entries: lane l gets v_rc[16 + (l&15)]
__device__ __forceinline__ float bcast_col(float v_rc, int lane) {
  float cross = xhalf(v_rc);
  return (lane < 16) ? cross : v_rc;
}

// fetch row ROW of a C-layout matrix, striped by column across all 32 lanes
template <int ROW>
__device__ __forceinline__ float get_row(const float* m, int h) {
  float own   = m[ROW & 7];
  float cross = xhalf(own);
  return (h == (ROW >> 3)) ? own : cross;
}

// G_rc * v for v in pos-layout (8 regs). Returns rc-layout scalar:
// lane<16 -> rowsum[lane], lane>=16 -> colsum[lane-16].
__device__ __forceinline__ float g_apply(const float* v, int lane) {
  float rb[8]; float cs = 0.f;
#pragma unroll
  for (int r = 0; r < 8; ++r) { rb[r] = half_rowsum(v[r]); cs += v[r]; }
  cs += xhalf(cs);
  float acc = 0.f;
#pragma unroll
  for (int r = 0; r < 8; ++r) {
    float cross = xhalf(rb[r]);                 // rowsum of the other half
    float w = (lane & 8) ? cross : rb[r];       // rowsum[(lane&15) row group]
    acc = ((lane & 7) == r) ? w : acc;
  }
  return (lane < 16) ? acc : cs;
}

__global__ __launch_bounds__(32) void qp_ipm_wave(const float* __restrict__ x,
                                                  float* __restrict__ out,
                                                  int batch) {
  const int lane = threadIdx.x;      // 0..31
  const int h    = lane >> 4;        // half-wave (row-offset bit)
  const int jl   = lane & 15;        // column index
  const int b    = blockIdx.x;
  if (b >= batch) return;

  const float* xb = x + (size_t)b * 256;

  // ---- state ----
  float z[8], p[8], sp[8], lp[8];    // primal z, cost p, pos slacks, pos duals
  float s_rc = 1.f, l_rc = 1.f;      // row/col slacks & duals (lane-mapped)
#pragma unroll
  for (int r = 0; r < 8; ++r) {
    p[r]  = -xb[(r + 8 * h) * 16 + jl];
    z[r]  = 0.f;
    sp[r] = 1.f;
    lp[r] = 1.f;
  }

#pragma unroll 1
  for (int it = 0; it < QP_IT; ++it) {
    // ---- r_dual = EPS*z + p + G^T lam ----
    const float lam_col = bcast_col(l_rc, lane);
    float rd[8];
#pragma unroll
    for (int r = 0; r < 8; ++r) {
      float lam_row = bcast_row(l_rc, r, h);
      rd[r] = QP_EPS * z[r] + p[r] + lam_row + lam_col - lp[r];
    }

    // ---- r_prim = G z + s - h ----
    const float rp_rc = g_apply(z, lane) + s_rc - 1.f;
    float rp[8];
#pragma unroll
    for (int r = 0; r < 8; ++r) rp[r] = -z[r] + sp[r];

    // ---- mu = mean(s .* lam) over 288 ----
    float acc = s_rc * l_rc;
#pragma unroll
    for (int r = 0; r < 8; ++r) acc += sp[r] * lp[r];
    const float mu  = wave_sum(acc) * (1.f / 288.f);
    const float smu = QP_SIG * mu;

    // ---- d = lam/s, rhs_lam = sigma*mu/s - lam + d*r_prim ----
    const float rcp_s_rc = frcp(s_rc);
    const float d_rc  = l_rc * rcp_s_rc;
    const float rl_rc = smu * rcp_s_rc - l_rc + d_rc * rp_rc;
    float dpos[8], rlp[8], rcp_sp[8];
#pragma unroll
    for (int r = 0; r < 8; ++r) {
      rcp_sp[r] = frcp(sp[r]);
      dpos[r]   = lp[r] * rcp_sp[r];
      rlp[r]    = smu * rcp_sp[r] - lp[r] + dpos[r] * rp[r];
    }

    // ---- rhs = -(r_dual + G^T rhs_lam) ----
    const float rl_col = bcast_col(rl_rc, lane);
    float rhs[8];
#pragma unroll
    for (int r = 0; r < 8; ++r) {
      float rl_row = bcast_row(rl_rc, r, h);
      rhs[r] = -(rd[r] + rl_row + rl_col - rlp[r]);
    }

    // ---- Woodbury: D = diag(EPS + d_pos), e = 1/D ----
    const float invd_rc = s_rc * frcp(l_rc);   // 1/d for row/col constraints
    float e[8], w[8], ainv_b[8], gr_b[8];
    float ce = 0.f, gc = 0.f;
#pragma unroll
    for (int r = 0; r < 8; ++r) {
      e[r] = frcp(QP_EPS + dpos[r]);
      float er = half_rowsum(e[r]);              // sum_j e[i][j]  (per-row)
      float gr = half_rowsum(e[r] * rhs[r]);     // g_row[i]
      ce += e[r];
      gc += e[r] * rhs[r];
      float a  = bcast_row(invd_rc, r, h) + er;  // a_i = 1/d_row + sum_j e
      ainv_b[r] = frcp(a);
      w[r]      = e[r] * ainv_b[r];              // W = A^{-1} E
      gr_b[r]   = gr;
    }
    ce += xhalf(ce); gc += xhalf(gc);            // col sums over all 16 rows
    const float invd_col = bcast_col(invd_rc, lane);
    const float bvec = invd_col + ce;            // b_j = 1/d_col + sum_i e

    // q = g_col - E^T A^{-1} g_row
    float qn = 0.f;
#pragma unroll
    for (int r = 0; r < 8; ++r) qn += w[r] * gr_b[r];
    qn += xhalf(qn);
    const float q = gc - qn;

    // ---- P = E^T * W via 4x V_WMMA_F32_16X16X4_F32 (K sliced by 4) ----
    v8f accv = {};
#pragma unroll
    for (int c = 0; c < 4; ++c) {
      // A operand: E^T (M=j, K=i). VGPR0: K=4c (half0) / K=4c+2 (half1); VGPR1: +1/+3
      v2f A, B;
      {
        float a0 = get_row<4 * 0 + 0>(e, h), t;  // placeholder; replaced below
        (void)a0; (void)t;
      }
      // manual unroll over the four K indices of this chunk
      const int K0 = 4 * c, K1 = 4 * c + 1, K2 = 4 * c + 2, K3 = 4 * c + 3;
      {
        float e0 = e[K0 & 7], e1 = e[K1 & 7], e2 = e[K2 & 7], e3 = e[K3 & 7];
        float x0 = xhalf(e0), x1 = xhalf(e1), x2 = xhalf(e2), x3 = xhalf(e3);
        float r0 = (h == (K0 >> 3)) ? e0 : x0;
        float r1 = (h == (K1 >> 3)) ? e1 : x1;
        float r2 = (h == (K2 >> 3)) ? e2 : x2;
        float r3 = (h == (K3 >> 3)) ? e3 : x3;
        A.x = h ? r2 : r0; A.y = h ? r3 : r1;
      }
      {
        float w0 = w[K0 & 7], w1 = w[K1 & 7], w2 = w[K2 & 7], w3 = w[K3 & 7];
        float x0 = xhalf(w0), x1 = xhalf(w1), x2 = xhalf(w2), x3 = xhalf(w3);
        float r0 = (h == (K0 >> 3)) ? w0 : x0;
        float r1 = (h == (K1 >> 3)) ? w1 : x1;
        float r2 = (h == (K2 >> 3)) ? w2 : x2;
        float r3 = (h == (K3 >> 3)) ? w3 : x3;
        B.x = h ? r2 : r0; B.y = h ? r3 : r1;
      }
      accv = __builtin_amdgcn_wmma_f32_16x16x4_f32(
          false, A, false, B, (short)0, accv, false, false);
    }

    // K = diag(b) - P   (Schur complement, 16x16 SPD, in C/D layout)
    float Kc[8];
#pragma unroll
    for (int r = 0; r < 8; ++r) {
      int rowj = r + 8 * h;
      Kc[r] = ((rowj == jl) ? bvec : 0.f) - accv[r];
    }

    // ---- solve K t_col = q via lane-parallel Gauss-Jordan (SPD) ----
    // rows striped across lanes; all pivot broadcasts are readlane (uniform)
    float krow[16], qb[16];
#pragma unroll
    for (int j = 0; j < 16; ++j) {
      float own = Kc[j & 7];
      float cr  = xhalf(own);
      krow[j] = (h == (j >> 3)) ? own : cr;     // row j striped by column
      qb[j]   = rdlane(q, j);                   // q[j] (uniform)
    }
#pragma unroll
    for (int pv = 0; pv < 16; ++pv) {
      float pinv = frcp(rdlane(krow[pv], pv));
      krow[pv] *= pinv; qb[pv] *= pinv;
#pragma unroll
      for (int rr = 0; rr < 16; ++rr) {
        if (rr == pv) continue;
        float f = rdlane(krow[rr], pv);         // uniform factor
        krow[rr] -= f * krow[pv];
        qb[rr]   -= f * qb[pv];
      }
    }
    float tc = 0.f;                             // t_col striped by lane
#pragma unroll
    for (int j = 0; j < 16; ++j) tc = (jl == j) ? qb[j] : tc;

    // ---- t_row = A^{-1}(g_row - E t_col); dz = e .* (rhs - t_row - t_col) ----
    float dz[8];
#pragma unroll
    for (int r = 0; r < 8; ++r) {
      float etc  = half_rowsum(e[r] * tc);
      float tr_b = (gr_b[r] - etc) * ainv_b[r];
      dz[r] = e[r] * (rhs[r] - tr_b - tc);
    }

    // ---- ds, dlam ----
    const float ds_rc = -rp_rc - g_apply(dz, lane);
    const float dl_rc = (smu - s_rc * l_rc - l_rc * ds_rc) * rcp_s_rc;
    float dsp[8], dlp[8];
#pragma unroll
    for (int r = 0; r < 8; ++r) {
      dsp[r] = -rp[r] + dz[r];
      dlp[r] = (smu - sp[r] * lp[r] - lp[r] * dsp[r]) * rcp_sp[r];
    }

    // ---- fraction-to-boundary step length ----
    float ratio = (ds_rc < 0.f) ? (s_rc * frcp(-ds_rc)) : 1e9f;
    {
      float t2 = (dl_rc < 0.f) ? (l_rc * frcp(-dl_rc)) : 1e9f;
      ratio = fminf(ratio, t2);
    }
#pragma unroll
    for (int r = 0; r < 8; ++r) {
      float t1 = (dsp[r] < 0.f) ? (sp[r] * frcp(-dsp[r])) : 1e9f;
      float t2 = (dlp[r] < 0.f) ? (lp[r] * frcp(-dlp[r])) : 1e9f;
      ratio = fminf(ratio, fminf(t1, t2));
    }
    const float alpha = 0.99f * fminf(1.f, wave_min(ratio));

    // ---- update ----
#pragma unroll
    for (int r = 0; r < 8; ++r) {
      z[r]  += alpha * dz[r];
      sp[r] += alpha * dsp[r];
      lp[r] += alpha * dlp[r];
    }
    s_rc += alpha * ds_rc;
    l_rc += alpha * dl_rc;
  }

  // ---- write result ----
  float* ob = out + (size_t)b * 256;
#pragma unroll
  for (int r = 0; r < 8; ++r) ob[(r + 8 * h) * 16 + jl] = z[r];
}

extern "C" void kernel_launch(void* const* d_in, const int* in_sizes, int n_in,
                              void* d_out, int out_size, void* d_ws, size_t ws_size,
                              hipStream_t stream) {
  (void)n_in; (void)out_size; (void)d_ws; (void)ws_size;
  const float* x = (const float*)d_in[0];
  float* out = (float*)d_out;
  const int batch = in_sizes[0] / 256;   // NSQ = 256
  qp_ipm_wave<<<dim3(batch), dim3(32), 0, stream>>>(x, out, batch);
}